// RGCN_27874337751212
// MI455X (gfx1250) — compile-verified
//
#include <hip/hip_runtime.h>

typedef float v2f __attribute__((ext_vector_type(2)));
typedef float v8f __attribute__((ext_vector_type(8)));

#define KC   32
#define LDA  36    // A-tile LDS row stride (floats): 128x32 tile, padded
#define LDB2 288   // B-tile pair-row stride (dwords): 128 cols * 2 + 32 pad

__device__ __forceinline__ void atomAddF(float* p, float v) {
  __hip_atomic_fetch_add(p, v, __ATOMIC_RELAXED, __HIP_MEMORY_SCOPE_AGENT);
}

// ---- gfx1250 async global->LDS copies (per-lane, tracked by ASYNCcnt) ----
__device__ __forceinline__ void async_load_b128(unsigned lds_off, const void* gptr) {
  asm volatile("global_load_async_to_lds_b128 %0, %1, off"
               :: "v"(lds_off), "v"((unsigned long long)(uintptr_t)gptr)
               : "memory");
}
__device__ __forceinline__ void async_load_b32(unsigned lds_off, const void* gptr) {
  asm volatile("global_load_async_to_lds_b32 %0, %1, off"
               :: "v"(lds_off), "v"((unsigned long long)(uintptr_t)gptr)
               : "memory");
}

__global__ void zero_kernel(float* __restrict__ p, long long n) {
  long long i = (long long)blockIdx.x * blockDim.x + threadIdx.x;
  if (i < n) p[i] = 0.0f;
}

__global__ void degree_kernel(const int* __restrict__ idx, float* __restrict__ deg, int E) {
  int i = blockIdx.x * blockDim.x + threadIdx.x;
  if (i < E) atomAddF(&deg[idx[i]], 1.0f);
}

__global__ void rsqrt_kernel(float* __restrict__ d, int n) {
  int i = blockIdx.x * blockDim.x + threadIdx.x;
  if (i < n) d[i] = rsqrtf(fmaxf(d[i], 1.0f));
}

// C[m][n] = rowscale[m] * sum_k A[m][k] * B[k][n]
// A: [M,K] row-major, B: [K,NN] row-major, K % 32 == 0, NN in {128,256}.
// Block: 256 threads = 8 waves. Block tile 128x128; wave = 16 rows x 128 cols.
// A tile is double-buffered; async copy of chunk c+1 overlaps compute of chunk c.
template <int NN>
__global__ __launch_bounds__(256) void gemm_rowscale_kernel(
    const float* __restrict__ A, const float* __restrict__ B,
    const float* __restrict__ rowscale, float* __restrict__ C,
    int M, int K)
{
  __shared__ float As[2][128 * LDA];      // 2 x 18432 B
  __shared__ float Bs[(KC / 2) * LDB2];   // 18432 B, pair-interleaved

  const int tid  = threadIdx.x;
  const int wave = tid >> 5;        // 0..7
  const int lane = tid & 31;
  const int half = lane >> 4;       // 0 or 1
  const int l16  = lane & 15;

  const int m0 = blockIdx.x * 128;
  const int n0 = blockIdx.y * 128;

  // ---- per-thread staging descriptors (loop-invariant; pointers advance by consts)
  // A: 4 b128/chunk; thread covers rows rA+32i, cols c4A..c4A+3.
  const int rA  = tid >> 3;
  const int c4A = (tid & 7) << 2;
  const unsigned aLds0 = (unsigned)(uintptr_t)(void*)&As[0][rA * LDA + c4A];
  const float* aP[4];
#pragma unroll
  for (int i = 0; i < 4; ++i) {
    int gr = m0 + rA + 32 * i;       // rows past M clamped: those acc rows never stored
    if (gr > M - 1) gr = M - 1;
    aP[i] = A + (long long)gr * K + c4A;
  }
  // B: 16 b32/chunk; thread covers rows r0+2i, col cB (pair-interleaved dest).
  const int r0 = tid >> 7;          // 0 or 1
  const int cB = tid & 127;
  const unsigned bLds0 = (unsigned)(uintptr_t)(void*)&Bs[cB * 2 + r0];
  const float* bP = B + (long long)r0 * NN + n0 + cB;

  // Prologue: stage A chunk 0 into buffer 0.
#pragma unroll
  for (int i = 0; i < 4; ++i) {
    async_load_b128(aLds0 + (unsigned)(i * 32 * LDA * 4), aP[i]);
    aP[i] += KC;
  }

  v8f acc[8] = {};
  const int numChunks = K / KC;

  for (int c = 0; c < numChunks; ++c) {
    __syncthreads();   // all waves done computing chunk c-1 -> Bs and As[(c+1)&1] free

    // B tile for chunk c (issued before A(c+1): async loads retire in order).
#pragma unroll
    for (int i = 0; i < 16; ++i)
      async_load_b32(bLds0 + (unsigned)(i * LDB2 * 4), bP + (long long)i * 2 * NN);
    bP += KC * NN;

    if (c + 1 < numChunks) {
      const unsigned abuf = (unsigned)((c + 1) & 1) * (128 * LDA * 4);
#pragma unroll
      for (int i = 0; i < 4; ++i) {
        async_load_b128(aLds0 + abuf + (unsigned)(i * 32 * LDA * 4), aP[i]);
        aP[i] += KC;
      }
      // pending (oldest->newest): A(c)=4, B(c)=16, A(c+1)=4 ; keep newest 4 in flight
      asm volatile("s_wait_asynccnt 0x4" ::: "memory");
    } else {
      asm volatile("s_wait_asynccnt 0x0" ::: "memory");
    }
    __syncthreads();

    const float* __restrict__ asrc = &As[c & 1][0];
#pragma unroll
    for (int kk = 0; kk < KC; kk += 4) {
      // A fragment (16x4 fp32): lanes0-15 -> K=kk,kk+1 ; lanes16-31 -> K=kk+2,kk+3
      const v2f a = *(const v2f*)&asrc[(16 * wave + l16) * LDA + kk + 2 * half];
      // B fragments: pair index (kk>>1)+half, adjacent dwords in interleaved layout.
      const int pbase = ((kk >> 1) + half) * LDB2 + l16 * 2;
      v2f bf[8];
#pragma unroll
      for (int t = 0; t < 8; ++t)
        bf[t] = *(const v2f*)&Bs[pbase + t * 32];
#pragma unroll
      for (int t = 0; t < 8; ++t)
        acc[t] = __builtin_amdgcn_wmma_f32_16x16x4_f32(
            false, a, false, bf[t], (short)0, acc[t], false, false);
    }
  }

  // Epilogue: C layout — VGPR r: lanes0-15 -> M=r, lanes16-31 -> M=r+8; N = lane&15.
#pragma unroll
  for (int t = 0; t < 8; ++t) {
#pragma unroll
    for (int r = 0; r < 8; ++r) {
      int row = m0 + 16 * wave + r + 8 * half;
      if (row < M) {
        int col = n0 + 16 * t + l16;
        C[(long long)row * NN + col] = acc[t][r] * rowscale[row];
      }
    }
  }
}

// One block per edge; blockDim = F/4. acc[dst] += msg[src] (float4 gather, f32 atomics).
__global__ void scatter_add_kernel(const float* __restrict__ msg,
                                   const int* __restrict__ src,
                                   const int* __restrict__ dst,
                                   float* __restrict__ acc, int F) {
  const int e = blockIdx.x;
  const int s = src[e];
  const int d = dst[e];
  const int f = threadIdx.x << 2;
  float4 v = *(const float4*)(msg + (long long)s * F + f);
  float* p = acc + (long long)d * F + f;
  atomAddF(p + 0, v.x);
  atomAddF(p + 1, v.y);
  atomAddF(p + 2, v.z);
  atomAddF(p + 3, v.w);
}

// x[i] = (x[i] * rs[i>>shiftF] + bias[i & (F-1)]), optional ReLU. In place.
__global__ void epilogue_kernel(float* __restrict__ x, const float* __restrict__ rs,
                                const float* __restrict__ bias, int shiftF, int doRelu,
                                long long total) {
  long long i = (long long)blockIdx.x * blockDim.x + threadIdx.x;
  if (i >= total) return;
  long long row = i >> shiftF;
  int f = (int)(i & ((1 << shiftF) - 1));
  float v = x[i] * rs[row] + bias[f];
  if (doRelu) v = fmaxf(v, 0.0f);
  x[i] = v;
}

static inline int cdiv_ll(long long a, long long b) { return (int)((a + b - 1) / b); }

static void launch_gemm(const float* A, const float* B, const float* rs, float* C,
                        int M, int N, int K, hipStream_t stream) {
  dim3 blk(256);
  if (N == 256) {
    gemm_rowscale_kernel<256><<<dim3(cdiv_ll(M, 128), 2), blk, 0, stream>>>(A, B, rs, C, M, K);
  } else {
    gemm_rowscale_kernel<128><<<dim3(cdiv_ll(M, 128), 1), blk, 0, stream>>>(A, B, rs, C, M, K);
  }
}

extern "C" void kernel_launch(void* const* d_in, const int* in_sizes, int n_in,
                              void* d_out, int out_size, void* d_ws, size_t ws_size,
                              hipStream_t stream) {
  (void)n_in; (void)out_size; (void)ws_size;

  const float* x_a   = (const float*)d_in[0];
  const float* x_b   = (const float*)d_in[1];
  const float* W1b   = (const float*)d_in[2];
  const float* b1b   = (const float*)d_in[3];
  const float* W1i   = (const float*)d_in[4];
  const float* b1i   = (const float*)d_in[5];
  const float* W2b   = (const float*)d_in[6];
  const float* b2b   = (const float*)d_in[7];
  const float* W2i   = (const float*)d_in[8];
  const float* b2i   = (const float*)d_in[9];
  const int* bel_src = (const int*)d_in[10];
  const int* bel_dst = (const int*)d_in[11];
  const int* inc_src = (const int*)d_in[12];
  const int* inc_dst = (const int*)d_in[13];

  const int HID  = in_sizes[3];          // 256
  const int OUT  = in_sizes[7];          // 128
  const int IN_F = in_sizes[2] / HID;    // 256
  const int N_A  = in_sizes[0] / IN_F;   // 50000
  const int N_B  = in_sizes[1] / IN_F;   // 10000
  const int E    = in_sizes[10];         // 300000
  const int shiftH = __builtin_ctz(HID);
  const int shiftO = __builtin_ctz(OUT);

  float* ws = (float*)d_ws;
  long long off = 0;
  float* rs_bel_src = ws + off; off += N_A;   // rsqrt(deg over bel_src) [A]
  float* rs_bel_dst = ws + off; off += N_B;   // rsqrt(deg over bel_dst) [B]
  float* rs_inc_src = ws + off; off += N_B;   // rsqrt(deg over inc_src) [B]
  float* rs_inc_dst = ws + off; off += N_A;   // rsqrt(deg over inc_dst) [A]
  float* t1    = ws + off; off += (long long)N_A * HID;  // x_a @ W1_bel (row-scaled)
  float* t2    = ws + off; off += (long long)N_B * HID;  // x_b @ W1_inc (row-scaled)
  float* acc_a = ws + off; off += (long long)N_A * HID;  // -> h_a (in place)
  float* acc_b = ws + off; off += (long long)N_B * HID;  // -> h_b (in place)
  float* t3 = t1;   // layer-2 message buffers reuse t1/t2
  float* t4 = t2;

  float* out_a = (float*)d_out;
  float* out_b = out_a + (long long)N_A * OUT;

  // 1) degrees -> rsqrt(max(deg,1)). The 4 rs arrays are contiguous.
  long long nrs = 2LL * (N_A + N_B);
  zero_kernel<<<cdiv_ll(nrs, 256), 256, 0, stream>>>(rs_bel_src, nrs);
  degree_kernel<<<cdiv_ll(E, 256), 256, 0, stream>>>(bel_src, rs_bel_src, E);
  degree_kernel<<<cdiv_ll(E, 256), 256, 0, stream>>>(bel_dst, rs_bel_dst, E);
  degree_kernel<<<cdiv_ll(E, 256), 256, 0, stream>>>(inc_src, rs_inc_src, E);
  degree_kernel<<<cdiv_ll(E, 256), 256, 0, stream>>>(inc_dst, rs_inc_dst, E);
  rsqrt_kernel<<<cdiv_ll(nrs, 256), 256, 0, stream>>>(rs_bel_src, (int)nrs);

  // 2) Layer-1 GEMMs: t = rowscale(src) * (X @ W1)
  launch_gemm(x_a, W1b, rs_bel_src, t1, N_A, HID, IN_F, stream);
  launch_gemm(x_b, W1i, rs_inc_src, t2, N_B, HID, IN_F, stream);

  // 3) Scatter-add over edges (acc_a, acc_b are contiguous -> one zero pass).
  long long nacc = (long long)(N_A + N_B) * HID;
  zero_kernel<<<cdiv_ll(nacc, 256), 256, 0, stream>>>(acc_a, nacc);
  scatter_add_kernel<<<E, HID / 4, 0, stream>>>(t1, bel_src, bel_dst, acc_b, HID);
  scatter_add_kernel<<<E, HID / 4, 0, stream>>>(t2, inc_src, inc_dst, acc_a, HID);

  // 4) h = relu(acc * rs_dst + b1)
  epilogue_kernel<<<cdiv_ll((long long)N_B * HID, 256), 256, 0, stream>>>(
      acc_b, rs_bel_dst, b1b, shiftH, 1, (long long)N_B * HID);
  epilogue_kernel<<<cdiv_ll((long long)N_A * HID, 256), 256, 0, stream>>>(
      acc_a, rs_inc_dst, b1i, shiftH, 1, (long long)N_A * HID);

  // 5) Layer-2 GEMMs
  launch_gemm(acc_a, W2b, rs_bel_src, t3, N_A, OUT, HID, stream);
  launch_gemm(acc_b, W2i, rs_inc_src, t4, N_B, OUT, HID, stream);

  // 6) Scatter into outputs, then scale + bias (no relu).
  long long nout = (long long)(N_A + N_B) * OUT;
  zero_kernel<<<cdiv_ll(nout, 256), 256, 0, stream>>>(out_a, nout);
  scatter_add_kernel<<<E, OUT / 4, 0, stream>>>(t3, bel_src, bel_dst, out_b, OUT);
  scatter_add_kernel<<<E, OUT / 4, 0, stream>>>(t4, inc_src, inc_dst, out_a, OUT);

  epilogue_kernel<<<cdiv_ll((long long)N_B * OUT, 256), 256, 0, stream>>>(
      out_b, rs_bel_dst, b2b, shiftO, 0, (long long)N_B * OUT);
  epilogue_kernel<<<cdiv_ll((long long)N_A * OUT, 256), 256, 0, stream>>>(
      out_a, rs_inc_dst, b2i, shiftO, 0, (long long)N_A * OUT);
}